// Mem2Seq_28449863369533
// MI455X (gfx1250) — compile-verified
//
#include <hip/hip_runtime.h>
#include <cstddef>

// Problem dims (compile-time, from reference)
#define VV 32000
#define DD 512
#define HH 3
#define BB 32
#define MM 512
#define TT 4
#define KK 1024                       // 2*DD
#define TAB ((size_t)VV * (size_t)DD) // elements per embedding table
#define CH 16                         // m-chunks for weighted-sum partials

typedef __attribute__((ext_vector_type(2))) float v2f;
typedef __attribute__((ext_vector_type(8))) float v8f;

// ---------------------------------------------------------------------------
// logits[b,m] = sum_t dot(C[story[b,m,t]], u[b])     one wave per (b,m)
// ---------------------------------------------------------------------------
__global__ __launch_bounds__(256) void attn_logits_kernel(
    const float* __restrict__ C, const float* __restrict__ u,
    const int* __restrict__ story, float* __restrict__ logits) {
  int wid  = blockIdx.x * 8 + (threadIdx.x >> 5);   // (b,m) pair, < BB*MM
  int lane = threadIdx.x & 31;
  int b = wid >> 9;                                  // wid / MM
  const int* st = story + (size_t)wid * TT;
  const float4* ub = (const float4*)(u + (size_t)b * DD);
  float4 u0 = ub[lane], u1 = ub[lane + 32], u2 = ub[lane + 64], u3 = ub[lane + 96];
  float acc = 0.f;
#pragma unroll
  for (int t = 0; t < TT; ++t) {
    const float4* row = (const float4*)(C + (size_t)st[t] * DD);
    float4 c0 = row[lane], c1 = row[lane + 32], c2 = row[lane + 64], c3 = row[lane + 96];
    acc += c0.x * u0.x + c0.y * u0.y + c0.z * u0.z + c0.w * u0.w;
    acc += c1.x * u1.x + c1.y * u1.y + c1.z * u1.z + c1.w * u1.w;
    acc += c2.x * u2.x + c2.y * u2.y + c2.z * u2.z + c2.w * u2.w;
    acc += c3.x * u3.x + c3.y * u3.y + c3.z * u3.z + c3.w * u3.w;
  }
#pragma unroll
  for (int off = 16; off > 0; off >>= 1) acc += __shfl_down(acc, off, 32);
  if (lane == 0) logits[wid] = acc;
}

// ---------------------------------------------------------------------------
// row softmax over M=512, one block per b
// ---------------------------------------------------------------------------
__global__ __launch_bounds__(256) void softmax_kernel(
    const float* __restrict__ logits, float* __restrict__ prob) {
  __shared__ float sm[256];
  int b = blockIdx.x, tid = threadIdx.x;
  const float* x = logits + (size_t)b * MM;
  float mx = -1e30f;
  for (int i = tid; i < MM; i += 256) mx = fmaxf(mx, x[i]);
  sm[tid] = mx; __syncthreads();
  for (int s = 128; s > 0; s >>= 1) { if (tid < s) sm[tid] = fmaxf(sm[tid], sm[tid + s]); __syncthreads(); }
  mx = sm[0]; __syncthreads();
  float sum = 0.f;
  for (int i = tid; i < MM; i += 256) { float e = expf(x[i] - mx); prob[(size_t)b * MM + i] = e; sum += e; }
  sm[tid] = sum; __syncthreads();
  for (int s = 128; s > 0; s >>= 1) { if (tid < s) sm[tid] += sm[tid + s]; __syncthreads(); }
  float inv = 1.f / sm[0];
  for (int i = tid; i < MM; i += 256) prob[(size_t)b * MM + i] *= inv;
}

// ---------------------------------------------------------------------------
// part[b,ch,d] = sum_{m in chunk} prob[b,m] * sum_t C[story[b,m,t]][d]
// grid (BB, CH), block 512 (one thread per d)
// ---------------------------------------------------------------------------
__global__ __launch_bounds__(512) void wsum_partial_kernel(
    const float* __restrict__ C, const float* __restrict__ prob,
    const int* __restrict__ story, float* __restrict__ part) {
  int b = blockIdx.x, ch = blockIdx.y, d = threadIdx.x;
  int mbase = ch * (MM / CH);
  const int*   st = story + ((size_t)b * MM + mbase) * TT;
  const float* pb = prob  + (size_t)b * MM + mbase;
  float acc = 0.f;
  for (int mm = 0; mm < MM / CH; ++mm) {
    float p = pb[mm];
    const int* s4 = st + mm * TT;
    float s = 0.f;
#pragma unroll
    for (int t = 0; t < TT; ++t) s += C[(size_t)s4[t] * DD + d];
    acc += p * s;
  }
  part[((size_t)b * CH + ch) * DD + d] = acc;
}

// out[b,d] (+)= sum_ch part[b,ch,d]
__global__ __launch_bounds__(256) void wsum_reduce_kernel(
    const float* __restrict__ part, float* __restrict__ out, int accumulate) {
  int i = blockIdx.x * 256 + threadIdx.x;       // < BB*DD
  if (i >= BB * DD) return;
  int b = i >> 9, d = i & (DD - 1);
  float s = 0.f;
#pragma unroll
  for (int ch = 0; ch < CH; ++ch) s += part[((size_t)b * CH + ch) * DD + d];
  if (accumulate) out[i] += s; else out[i] = s;
}

// ---------------------------------------------------------------------------
// small elementwise helpers
// ---------------------------------------------------------------------------
__global__ __launch_bounds__(256) void zero_kernel(float* p, int n) {
  int i = blockIdx.x * 256 + threadIdx.x; if (i < n) p[i] = 0.f;
}
__global__ __launch_bounds__(256) void copy_kernel(const float* __restrict__ s, float* __restrict__ d, int n) {
  int i = blockIdx.x * 256 + threadIdx.x; if (i < n) d[i] = s[i];
}
__global__ __launch_bounds__(256) void add_kernel(float* __restrict__ d, const float* __restrict__ s, int n) {
  int i = blockIdx.x * 256 + threadIdx.x; if (i < n) d[i] += s[i];
}
__global__ __launch_bounds__(256) void gather_x_kernel(
    const float* __restrict__ C0, const int* __restrict__ q, float* __restrict__ x) {
  int i = blockIdx.x * 256 + threadIdx.x;       // < BB*DD
  if (i >= BB * DD) return;
  int b = i >> 9, d = i & (DD - 1);
  x[i] = C0[(size_t)q[b] * DD + d];
}
__global__ __launch_bounds__(256) void build_cat_kernel(
    const float* __restrict__ hid, const float* __restrict__ ok, float* __restrict__ cat) {
  int i = blockIdx.x * 256 + threadIdx.x;       // < BB*KK
  if (i >= BB * KK) return;
  int b = i >> 10, k = i & (KK - 1);
  cat[i] = (k < DD) ? hid[(size_t)b * DD + k] : ok[(size_t)b * DD + (k - DD)];
}

// ---------------------------------------------------------------------------
// GRU gate GEMV: out[b,j] = dot(X[b,:], W[j,:]) + bias[j]   one wave per (b,j)
// W: [3D, D] row-major
// ---------------------------------------------------------------------------
__global__ __launch_bounds__(256) void gemv3d_kernel(
    const float* __restrict__ W, const float* __restrict__ bias,
    const float* __restrict__ X, float* __restrict__ out) {
  int wid  = blockIdx.x * 8 + (threadIdx.x >> 5);   // < BB*3*DD
  int lane = threadIdx.x & 31;
  int b = wid / (3 * DD), j = wid % (3 * DD);
  const float4* wr = (const float4*)(W + (size_t)j * DD);
  const float4* xr = (const float4*)(X + (size_t)b * DD);
  float acc = 0.f;
#pragma unroll
  for (int i = 0; i < 4; ++i) {
    float4 w4 = wr[lane + 32 * i], x4 = xr[lane + 32 * i];
    acc += w4.x * x4.x + w4.y * x4.y + w4.z * x4.z + w4.w * x4.w;
  }
#pragma unroll
  for (int off = 16; off > 0; off >>= 1) acc += __shfl_down(acc, off, 32);
  if (lane == 0) out[(size_t)b * (3 * DD) + j] = acc + bias[j];
}

__global__ __launch_bounds__(256) void gru_elem_kernel(
    const float* __restrict__ gi, const float* __restrict__ gh,
    const float* __restrict__ h0, float* __restrict__ hidden,
    float* __restrict__ hidden_out) {
  int i = blockIdx.x * 256 + threadIdx.x;       // < BB*DD
  if (i >= BB * DD) return;
  int b = i >> 9, d = i & (DD - 1);
  const float* gib = gi + (size_t)b * (3 * DD);
  const float* ghb = gh + (size_t)b * (3 * DD);
  float r = 1.f / (1.f + expf(-(gib[d] + ghb[d])));
  float z = 1.f / (1.f + expf(-(gib[DD + d] + ghb[DD + d])));
  float n = tanhf(gib[2 * DD + d] + r * ghb[2 * DD + d]);
  float h = (1.f - z) * n + z * h0[i];
  hidden[i] = h;
  hidden_out[i] = h;
}

// ---------------------------------------------------------------------------
// p_vocab GEMM via V_WMMA_F32_16X16X4_F32 (full f32 precision).
// out[b,n] = sum_k cat[b,k] * W1[n,k] + W1b[n];  B=32 (2 M-tiles), K=1024, N=32000
// One wave per 16-wide N tile, both M tiles per wave -> W1 streamed exactly once.
// 250 blocks * 8 waves = 2000 N tiles (no divergence; EXEC all ones for WMMA).
// ---------------------------------------------------------------------------
__global__ __launch_bounds__(256) void pvocab_wmma_kernel(
    const float* __restrict__ cat, const float* __restrict__ W1,
    const float* __restrict__ W1b, float* __restrict__ out) {
  int nt   = blockIdx.x * 8 + (threadIdx.x >> 5);   // N tile index, < 2000
  int lane = threadIdx.x & 31;
  int half = lane >> 4;          // K sub-group select
  int l15  = lane & 15;          // row/col within tile

  // A fragment lanes: lane<16 -> M=l15, K pair {0,1}; lane>=16 -> K pair {2,3}
  const float* a0row = cat + (size_t)l15 * KK        + half * 2;   // M tile 0 (b 0..15)
  const float* a1row = cat + (size_t)(16 + l15) * KK + half * 2;   // M tile 1 (b 16..31)
  // B fragment lanes: lane<16 -> N=l15, K pair {0,1}; lane>=16 -> K pair {2,3}
  const float* w1row = W1 + (size_t)(nt * 16 + l15) * KK + half * 2;

  v8f c0 = {}; v8f c1 = {};
  for (int kc = 0; kc < KK; kc += 128) {
    if (kc + 128 < KK) __builtin_prefetch(w1row + kc + 128, 0, 1);  // global_prefetch_b8
#pragma unroll 8
    for (int k4 = kc; k4 < kc + 128; k4 += 4) {
      v2f bf = *(const v2f*)(w1row + k4);
      v2f a0 = *(const v2f*)(a0row + k4);
      v2f a1 = *(const v2f*)(a1row + k4);
      c0 = __builtin_amdgcn_wmma_f32_16x16x4_f32(false, a0, false, bf, (short)0, c0, false, false);
      c1 = __builtin_amdgcn_wmma_f32_16x16x4_f32(false, a1, false, bf, (short)0, c1, false, false);
    }
  }

  // C/D layout: VGPR r -> M=r (lanes 0-15) / M=r+8 (lanes 16-31); N=l15
  int n    = nt * 16 + l15;
  float bv = W1b[n];
  int mb   = half * 8;
#pragma unroll
  for (int r = 0; r < 8; ++r) {
    out[(size_t)(mb + r) * VV + n]      = c0[r] + bv;   // rows 0..15
    out[(size_t)(16 + mb + r) * VV + n] = c1[r] + bv;   // rows 16..31
  }
}

// ---------------------------------------------------------------------------
// host orchestration
// ---------------------------------------------------------------------------
extern "C" void kernel_launch(void* const* d_in, const int* in_sizes, int n_in,
                              void* d_out, int out_size, void* d_ws, size_t ws_size,
                              hipStream_t stream) {
  const int*   story = (const int*)d_in[0];
  const int*   encq  = (const int*)d_in[1];
  const float* C_enc = (const float*)d_in[2];
  const float* C_dec = (const float*)d_in[3];
  const float* w_ih  = (const float*)d_in[4];
  const float* w_hh  = (const float*)d_in[5];
  const float* b_ih  = (const float*)d_in[6];
  const float* b_hh  = (const float*)d_in[7];
  const float* W1    = (const float*)d_in[8];
  const float* W1b   = (const float*)d_in[9];

  float* out        = (float*)d_out;
  float* p_ptr_out  = out;                       // [32,512]
  float* p_voc_out  = out + BB * MM;             // [32,32000]
  float* hidden_out = out + BB * MM + BB * VV;   // [1,32,512]

  float* ws     = (float*)d_ws;
  float* u      = ws;                  // BB*DD
  float* logits = u + BB * DD;         // BB*MM
  float* prob   = logits + BB * MM;    // BB*MM
  float* part   = prob + BB * MM;      // BB*CH*DD
  float* o_k    = part + BB * CH * DD; // BB*DD
  float* uk     = o_k + BB * DD;       // BB*DD
  float* hid    = uk + BB * DD;        // BB*DD
  float* x      = hid + BB * DD;       // BB*DD
  float* gi     = x + BB * DD;         // BB*3*DD
  float* gh     = gi + BB * 3 * DD;    // BB*3*DD
  float* cat    = gh + BB * 3 * DD;    // BB*KK

  const dim3 wsGrid(BB, CH);

  // ---- encoder: u = 0; 3 hops of fused attention ----
  zero_kernel<<<(BB * DD) / 256, 256, 0, stream>>>(u, BB * DD);
  for (int h = 0; h < HH; ++h) {
    attn_logits_kernel<<<(BB * MM) / 8, 256, 0, stream>>>(C_enc + (size_t)h * TAB, u, story, logits);
    softmax_kernel<<<BB, 256, 0, stream>>>(logits, prob);
    wsum_partial_kernel<<<wsGrid, DD, 0, stream>>>(C_enc + (size_t)(h + 1) * TAB, prob, story, part);
    wsum_reduce_kernel<<<(BB * DD) / 256, 256, 0, stream>>>(part, u, 1);
  }

  // ---- GRU step ----
  gather_x_kernel<<<(BB * DD) / 256, 256, 0, stream>>>(C_dec, encq, x);
  gemv3d_kernel<<<(BB * 3 * DD) / 8, 256, 0, stream>>>(w_ih, b_ih, x, gi);
  gemv3d_kernel<<<(BB * 3 * DD) / 8, 256, 0, stream>>>(w_hh, b_hh, u, gh);
  gru_elem_kernel<<<(BB * DD) / 256, 256, 0, stream>>>(gi, gh, u, hid, hidden_out);
  copy_kernel<<<(BB * DD) / 256, 256, 0, stream>>>(hid, uk, BB * DD);

  // ---- decoder hop 0 ----
  attn_logits_kernel<<<(BB * MM) / 8, 256, 0, stream>>>(C_dec, uk, story, logits);
  softmax_kernel<<<BB, 256, 0, stream>>>(logits, prob);
  wsum_partial_kernel<<<wsGrid, DD, 0, stream>>>(C_dec + TAB, prob, story, part);
  wsum_reduce_kernel<<<(BB * DD) / 256, 256, 0, stream>>>(part, o_k, 0);
  build_cat_kernel<<<(BB * KK) / 256, 256, 0, stream>>>(hid, o_k, cat);
  pvocab_wmma_kernel<<<VV / 128, 256, 0, stream>>>(cat, W1, W1b, p_voc_out);
  add_kernel<<<(BB * DD) / 256, 256, 0, stream>>>(uk, o_k, BB * DD);

  // ---- decoder hop 1 ----
  attn_logits_kernel<<<(BB * MM) / 8, 256, 0, stream>>>(C_dec + TAB, uk, story, logits);
  softmax_kernel<<<BB, 256, 0, stream>>>(logits, prob);
  wsum_partial_kernel<<<wsGrid, DD, 0, stream>>>(C_dec + 2 * TAB, prob, story, part);
  wsum_reduce_kernel<<<(BB * DD) / 256, 256, 0, stream>>>(part, o_k, 0);
  add_kernel<<<(BB * DD) / 256, 256, 0, stream>>>(uk, o_k, BB * DD);

  // ---- decoder hop 2: p_ptr = raw logits (softmax/o_k/uk are dead work) ----
  attn_logits_kernel<<<(BB * MM) / 8, 256, 0, stream>>>(C_dec + 2 * TAB, uk, story, p_ptr_out);
}